// CCLoss_46205258170952
// MI455X (gfx1250) — compile-verified
//
#include <hip/hip_runtime.h>

// LNCC (local normalized cross-correlation) loss, 4x4x4 box window, pad=1.
// Inputs: d_in[0]=fix_img f32[160*192*192], d_in[1]=reg_img f32[160*192*192],
//         d_in[2]=deform_space (UNUSED by the reference computation).
// Output: d_out[0] = -mean(cc) over the 159x191x191 valid output volume.

typedef __attribute__((ext_vector_type(2))) float v2f;
typedef __attribute__((ext_vector_type(8))) float v8f;

#define D_IN  160
#define H_IN  192
#define W_IN  192
#define D_OUT 159
#define H_OUT 191
#define W_OUT 191
#define TILE  16
#define HALO  19   // input footprint of a 16-wide output tile (o-1 .. o+17)
#define PADT  20   // halo padded to 20 so K splits into 5 chunks of 4
#define DC    8    // output D-planes per workgroup

__global__ void CC_zero_out(float* out) { out[0] = 0.0f; }

__global__ __launch_bounds__(32)
void CC_loss_kernel(const float* __restrict__ gI,
                    const float* __restrict__ gJ,
                    float* __restrict__ out) {
    // 4-deep ring of raw I/J halo planes + 5-channel D-summed plane + W-summed T
    __shared__ float ringI[4][PADT * PADT];
    __shared__ float ringJ[4][PADT * PADT];
    __shared__ float sU[5][PADT * PADT];   // D-direction box sums, 5 moments
    __shared__ float sT[5][PADT * TILE];   // + W-direction box sums

    const int lane = threadIdx.x;          // wave32: one wave per workgroup
    const int m    = lane & 15;
    const int hi   = lane >> 4;            // 0 = lanes 0-15, 1 = lanes 16-31
    const int h0   = blockIdx.x * TILE;
    const int w0   = blockIdx.y * TILE;
    const int z0   = blockIdx.z * DC;
    const int dend = min(z0 + DC - 1, D_OUT - 1);

    float partial = 0.0f;

    for (int din = z0 - 1; din <= dend + 2; ++din) {
        const int slot = din & 3;

        // ---- stage one 19x19 halo plane of I and J (zero padded to 20x20) ----
        for (int t = lane; t < PADT * PADT; t += 32) {
            int th = t / PADT, tw = t % PADT;
            int h = h0 - 1 + th, w = w0 - 1 + tw;
            bool v = (th < HALO) && (tw < HALO) &&
                     ((unsigned)h < H_IN) && ((unsigned)w < W_IN) &&
                     ((unsigned)din < D_IN);
            float a = 0.0f, b = 0.0f;
            if (v) {
                size_t idx = ((size_t)din * H_IN + (size_t)h) * W_IN + (size_t)w;
                a = gI[idx];
                b = gJ[idx];
            }
            ringI[slot][t] = a;
            ringJ[slot][t] = b;
        }
        __syncthreads();

        const int dout = din - 2;                 // ring now holds dout-1..dout+2
        if (dout >= z0 && dout <= dend) {         // wave-uniform branch
            // ---- D-direction box sum fused with the 5 moment channels ----
            for (int t = lane; t < PADT * PADT; t += 32) {
                float i0 = ringI[0][t], i1 = ringI[1][t];
                float i2 = ringI[2][t], i3 = ringI[3][t];
                float j0 = ringJ[0][t], j1 = ringJ[1][t];
                float j2 = ringJ[2][t], j3 = ringJ[3][t];
                sU[0][t] = i0 + i1 + i2 + i3;
                sU[1][t] = j0 + j1 + j2 + j3;
                sU[2][t] = i0*i0 + i1*i1 + i2*i2 + i3*i3;
                sU[3][t] = j0*j0 + j1*j1 + j2*j2 + j3*j3;
                sU[4][t] = i0*j0 + i1*j1 + i2*j2 + i3*j3;
            }
            __syncthreads();

            // ---- W-direction box sum: T[c][th][j] = sum_{k=0..3} U[c][th][j+k] ----
            for (int t = lane; t < 5 * PADT * TILE; t += 32) {
                int c  = t / (PADT * TILE);
                int r  = t % (PADT * TILE);
                int th = r / TILE, j = r % TILE;
                const float* u = &sU[c][th * PADT + j];
                sT[c][r] = u[0] + u[1] + u[2] + u[3];
            }
            __syncthreads();

            // ---- H-direction box sum as banded matmul on the tensor core:
            //      Q(16x16) = Bh(16x20) x T(20x16), Bh[m][k] = (m <= k <= m+3).
            //      A layout (16x4 f32): VGPR v, lane L -> M=L&15, K=kb+2*(L>>4)+v
            //      B layout (4x16 f32): VGPR v, lane L -> N=L&15, K=kb+2*(L>>4)+v
            v8f Q[5];
#pragma unroll
            for (int c = 0; c < 5; ++c) {
                v8f acc = {0.f, 0.f, 0.f, 0.f, 0.f, 0.f, 0.f, 0.f};
#pragma unroll
                for (int kb = 0; kb < PADT; kb += 4) {
                    int k0 = kb + 2 * hi;
                    v2f A, B;
                    A.x = (k0     >= m && k0     <= m + 3) ? 1.0f : 0.0f;
                    A.y = (k0 + 1 >= m && k0 + 1 <= m + 3) ? 1.0f : 0.0f;
                    B.x = sT[c][(k0    ) * TILE + m];
                    B.y = sT[c][(k0 + 1) * TILE + m];
                    acc = __builtin_amdgcn_wmma_f32_16x16x4_f32(
                        /*neg_a=*/false, A, /*neg_b=*/false, B,
                        /*c_mod=*/(short)0, acc,
                        /*reuse_a=*/false, /*reuse_b=*/false);
                }
                Q[c] = acc;
            }

            // ---- pointwise LNCC on the WMMA accumulator layout ----
            const float wrec = 1.0f / 64.0f;
            v8f uI    = Q[0] * wrec;
            v8f uJ    = Q[1] * wrec;
            v8f cross = Q[4] - uJ * Q[0] - uI * Q[1] + uI * uJ * 64.0f;
            v8f Ivar  = Q[2] - 2.0f * uI * Q[0] + uI * uI * 64.0f;
            v8f Jvar  = Q[3] - 2.0f * uJ * Q[1] + uJ * uJ * 64.0f;
            v8f cc    = (cross * cross) / (Ivar * Jvar + 1e-5f);

            // C/D layout: element i of lane L -> (M = i + 8*(L>>4), N = L&15)
#pragma unroll
            for (int i = 0; i < 8; ++i) {
                int ho = h0 + i + 8 * hi;
                int wo = w0 + m;
                partial += (ho < H_OUT && wo < W_OUT) ? cc[i] : 0.0f;
            }
        }
        __syncthreads();   // ring slot / sT reuse fence for next iteration
    }

    // ---- wave32 reduction + single atomic per wave ----
    for (int off = 16; off > 0; off >>= 1)
        partial += __shfl_xor(partial, off, 32);
    if (lane == 0) {
        const float scale =
            -1.0f / ((float)D_OUT * (float)H_OUT * (float)W_OUT);
        atomicAdd(out, partial * scale);
    }
}

extern "C" void kernel_launch(void* const* d_in, const int* in_sizes, int n_in,
                              void* d_out, int out_size, void* d_ws, size_t ws_size,
                              hipStream_t stream) {
    const float* fix_img = (const float*)d_in[0];
    const float* reg_img = (const float*)d_in[1];
    // d_in[2] (deform_space) is unused by the reference computation.
    float* out = (float*)d_out;

    CC_zero_out<<<dim3(1), dim3(1), 0, stream>>>(out);

    dim3 grid((H_OUT + TILE - 1) / TILE,   // 12
              (W_OUT + TILE - 1) / TILE,   // 12
              (D_OUT + DC - 1) / DC);      // 20
    dim3 block(32);
    CC_loss_kernel<<<grid, block, 0, stream>>>(fix_img, reg_img, out);
}